// MCPBRNN_Generic_PETconstraint_constantoutput_variableLoss_BYPASSM0_53695681134734
// MI455X (gfx1250) — compile-verified
//
#include <hip/hip_runtime.h>
#include <stdint.h>

#define T_N       50000
#define SPIN      365
#define TRAIN     40000
#define ML_C      2.9086f
#define SL_C      1.898f
#define CHUNK     512

// ---------- helpers ----------
__device__ __forceinline__ uint32_t lds_off(const void* p) {
  // generic pointer from __shared__ : low 32 bits == LDS byte offset
  return (uint32_t)(uintptr_t)p;
}

// CDNA5 async copy: 32 lanes x 16B per instruction = 512B/instr, ASYNCcnt-tracked
__device__ __forceinline__ void async_chunk_to_lds(const float* gbase, long gfirst,
                                                   long gmax_first, float* ldsbase,
                                                   int nElem, int lane) {
  uint32_t l0 = lds_off(ldsbase);
#pragma unroll
  for (int k = 0; k < nElem / 128; ++k) {
    int e  = k * 128 + lane * 4;            // float index inside chunk
    long g = gfirst + e;
    if (g > gmax_first) g = gmax_first;     // clamp tail (data unused, stay in-bounds)
    asm volatile("global_load_async_to_lds_b128 %0, %1, off"
                 :: "v"(l0 + (uint32_t)(e * 4)), "v"(gbase + g)
                 : "memory");
  }
}

__device__ __forceinline__ void wait_async0() {
  asm volatile("s_wait_asynccnt 0" ::: "memory");
}

// ---------- K0: std(y_obs[365:40000], ddof=1) -> ws[0] ----------
__global__ __launch_bounds__(1024) void k0_obsstd(const float* __restrict__ y,
                                                  float* __restrict__ ws) {
  __shared__ double s_sum[1024];
  __shared__ double s_sq[1024];
  double s = 0.0, q = 0.0;
  for (int i = SPIN + (int)threadIdx.x; i < TRAIN; i += 1024) {
    double v = (double)y[i];
    s += v; q += v * v;
  }
  s_sum[threadIdx.x] = s; s_sq[threadIdx.x] = q;
  __syncthreads();
  for (int off = 512; off > 0; off >>= 1) {
    if ((int)threadIdx.x < off) {
      s_sum[threadIdx.x] += s_sum[threadIdx.x + off];
      s_sq[threadIdx.x]  += s_sq[threadIdx.x + off];
    }
    __syncthreads();
  }
  if (threadIdx.x == 0) {
    const double m = (double)(TRAIN - SPIN);
    double mean = s_sum[0] / m;
    double var  = (s_sq[0] - mean * mean * m) / (m - 1.0);
    ws[0] = (float)sqrt(var);
  }
}

// ---------- K1: parallel precompute ol + constant/masked outputs ----------
__global__ void k1_ol(const float* __restrict__ x,
                      const float* __restrict__ wom, const float* __restrict__ wlm,
                      const float* __restrict__ wfm, const float* __restrict__ b0,
                      const float* __restrict__ b2, const int* __restrict__ tlp,
                      const float* __restrict__ ws, float* __restrict__ out) {
  int t = blockIdx.x * blockDim.x + threadIdx.x;
  if (t >= T_N) return;
  float eo = __expf(wom[0]), el = __expf(wlm[0]), ef = __expf(wfm[0]);
  float den = eo + el + ef;
  float oo = eo / den, ol1 = el / den;
  int tl = *tlp;
  bool m = (t >= tl);
  float u2  = x[2 * t + 1];
  float ol3 = b0[0] + (u2 - ML_C) * (1.0f / SL_C) * b2[0];
  float ol  = ol1 / (1.0f + __expf(-ol3));
  float ostd = ws[0];
  out[4 * T_N + t]        = 0.0f;               // bp_n
  out[5 * T_N + t]        = 0.0f;               // g_ib
  out[6 * T_N + t]        = m ? oo : 0.0f;      // g_oo
  out[7 * T_N + t]        = m ? ol : 0.0f;      // g_ol (also feeds serial scan)
  out[10 * T_N + 2*t + 1] = m ? ostd : 0.0f;    // h_nout col 1
  out[12 * T_N + t]       = m ? ostd : 0.0f;    // obs_std
}

// ---------- K2: serial scan (1 wave; lane0 computes, wave prefetches) ----------
__global__ __launch_bounds__(32) void k2_scan(const float* __restrict__ x,
                                              const float* __restrict__ wom,
                                              const float* __restrict__ wlm,
                                              const float* __restrict__ wfm,
                                              const int* __restrict__ tlp,
                                              float* __restrict__ out) {
  __shared__ float xbuf[2][CHUNK * 2];   // (u1,u2) interleaved, matches x layout
  __shared__ float olbuf[2][CHUNK];
  const int lane = threadIdx.x;
  const float* ol_arr = out + 7 * (long)T_N;   // masked ol written by K1
  float* c_out = out + 1 * (long)T_N;          // c_n

  float eo = __expf(wom[0]), el = __expf(wlm[0]), ef = __expf(wfm[0]);
  float oo = eo / (eo + el + ef);
  const int tl = *tlp;
  const int nchunks = (T_N + CHUNK - 1) / CHUNK;

  // prime chunk 0
  async_chunk_to_lds(x, 0, 2L * T_N - 4, &xbuf[0][0], CHUNK * 2, lane);
  async_chunk_to_lds(ol_arr, 0, (long)T_N - 4, &olbuf[0][0], CHUNK, lane);
  wait_async0();

  float c = 0.0f;
  int t = 0;
  for (int ch = 0; ch < nchunks; ++ch) {
    if (ch + 1 < nchunks) {
      long nb = (long)(ch + 1) * CHUNK;
      int b = (ch + 1) & 1;
      async_chunk_to_lds(x, nb * 2, 2L * T_N - 4, &xbuf[b][0], CHUNK * 2, lane);
      async_chunk_to_lds(ol_arr, nb, (long)T_N - 4, &olbuf[b][0], CHUNK, lane);
    }
    int buf = ch & 1;
    int cnt = T_N - ch * CHUNK; if (cnt > CHUNK) cnt = CHUNK;
    if (lane == 0) {
      for (int i = 0; i < cnt; ++i) {
        float u1 = xbuf[buf][2 * i];
        float u2 = xbuf[buf][2 * i + 1];
        float ol = olbuf[buf][i];
        c_out[t + i] = c;                               // c0 (pre-update) == c_n
        // ---- dependency chain ----
        float rc   = __builtin_amdgcn_rcpf(c > 0.0f ? c : 1.0f);
        float d    = ol - u2 * rc;
        float eluD = d > 0.0f ? d : (__expf(d) - 1.0f);
        float pre  = c > 0.0f ? (ol - eluD) : ol;
        float f    = 1.0f - oo - pre; f = f > 0.0f ? f : 0.0f;
        float c1   = __builtin_fmaf(f, c, u1);
        c = ((t + i) >= tl) ? c1 : c;
      }
    }
    t += cnt;
    wait_async0();   // next buffer ready before swap
  }
}

// ---------- K3: parallel finalize from (c_n, ol, u2) ----------
__global__ void k3_fin(const float* __restrict__ x,
                       const float* __restrict__ wom, const float* __restrict__ wlm,
                       const float* __restrict__ wfm, const int* __restrict__ tlp,
                       float* __restrict__ out) {
  int t = blockIdx.x * blockDim.x + threadIdx.x;
  if (t >= T_N) return;
  float eo = __expf(wom[0]), el = __expf(wlm[0]), ef = __expf(wfm[0]);
  float oo = eo / (eo + el + ef);
  int tl = *tlp;
  bool m = (t >= tl);
  float c  = out[1 * T_N + t];     // c0 at step t (written by K2)
  float ol = out[7 * T_N + t];     // masked ol
  float u2 = x[2 * t + 1];
  float rc   = __builtin_amdgcn_rcpf(c > 0.0f ? c : 1.0f);
  float d    = ol - u2 * rc;
  float eluD = d > 0.0f ? d : (__expf(d) - 1.0f);
  float pre  = c > 0.0f ? (ol - eluD) : ol;
  float f    = fmaxf(0.0f, 1.0f - oo - pre);
  float olc  = fmaxf(0.0f, pre);
  float h    = m ? oo * c : 0.0f;
  out[0 * T_N + t]       = h;                     // h_n
  out[1 * T_N + t]       = m ? c : 0.0f;          // c_n (apply mask)
  out[2 * T_N + t]       = m ? ol * c : 0.0f;     // l_n
  out[3 * T_N + t]       = m ? olc * c : 0.0f;    // lc_n
  out[8 * T_N + t]       = m ? olc : 0.0f;        // g_olc
  out[9 * T_N + t]       = m ? f : 0.0f;          // g_f
  out[10 * T_N + 2 * t]  = h;                     // h_nout col 0
}

extern "C" void kernel_launch(void* const* d_in, const int* in_sizes, int n_in,
                              void* d_out, int out_size, void* d_ws, size_t ws_size,
                              hipStream_t stream) {
  const float* x    = (const float*)d_in[0];
  const float* yobs = (const float*)d_in[1];
  const float* wom  = (const float*)d_in[2];
  const float* wlm  = (const float*)d_in[3];
  const float* wfm  = (const float*)d_in[4];
  const float* b0   = (const float*)d_in[5];
  const float* b2   = (const float*)d_in[6];
  // d_in[7] = theltaC (unused), d_in[8] = epoch (unused)
  const int*   tl   = (const int*)d_in[9];
  float* out = (float*)d_out;
  float* ws  = (float*)d_ws;

  k0_obsstd<<<1, 1024, 0, stream>>>(yobs, ws);
  k1_ol<<<(T_N + 255) / 256, 256, 0, stream>>>(x, wom, wlm, wfm, b0, b2, tl, ws, out);
  k2_scan<<<1, 32, 0, stream>>>(x, wom, wlm, wfm, tl, out);
  k3_fin<<<(T_N + 255) / 256, 256, 0, stream>>>(x, wom, wlm, wfm, tl, out);
}